// SpatialTransformer_24352464569131
// MI455X (gfx1250) — compile-verified
//
#include <hip/hip_runtime.h>

typedef float v4f __attribute__((ext_vector_type(4)));

// Problem shape (fixed by the reference setup_inputs).
constexpr int B = 2, C = 32, D = 12, H = 192, W = 384;
constexpr int HW = H * W;

// One block per (b, h); 384 threads = 12 wave32s.
// Thread map: q = t % 96 -> column quad (w = 4q..4q+3); cg = t / 96 -> group
// of 8 channels. Right-image rows for this (b,h) are staged into LDS with
// CDNA5 async global->LDS b128 copies; the left-feature broadcast streams out
// underneath the async DMA; then the disparity warp gathers from LDS and
// writes b128 non-temporal stores.
__global__ __launch_bounds__(384) void disparity_warp_kernel(
    const float* __restrict__ left,
    const float* __restrict__ right,
    const float* __restrict__ disp,
    float* __restrict__ warped,
    float* __restrict__ leftout)
{
    __shared__ float smem[C * W];   // 48 KB: right[b, c, h, w] -> smem[c*W + w]

    const int bh = blockIdx.x;      // 0 .. B*H-1
    const int b  = bh / H;
    const int h  = bh % H;
    const int t  = threadIdx.x;
    const int q  = t % 96;          // column quad index
    const int cg = t / 96;          // channel group 0..3 (8 channels each)
    const int wb = q * 4;           // first of 4 owned columns

    // ---- Issue async global->LDS copies of the 32 right rows ----
    // Each lane moves 16 B; one row = 96 lanes; 384 threads cover 4 rows per
    // iteration; 8 iterations cover all 32 channels.
#pragma unroll
    for (int it = 0; it < 8; ++it) {
        const int c = cg + it * 4;
        const unsigned long long g = (unsigned long long)
            (right + ((size_t)(b * C + c) * H + h) * W + wb);
        const unsigned l = (unsigned)(size_t)(&smem[c * W + wb]);
        asm volatile("global_load_async_to_lds_b128 %0, %1, off"
                     :: "v"(l), "v"(g) : "memory");
    }

    // ---- Left-feature broadcast: overlaps the async staging (no LDS dep) ----
    {
        const size_t lbase = ((size_t)(b * C + cg * 8) * H + h) * W + wb;
        const size_t obase = (size_t)b * C * D * HW
                           + (size_t)(cg * 8) * D * HW
                           + (size_t)h * W + wb;
#pragma unroll
        for (int k = 0; k < 8; ++k) {
            const v4f lv = __builtin_nontemporal_load(
                (const v4f*)(left + lbase + (size_t)k * HW));
            size_t o = obase + (size_t)k * (D * HW);
#pragma unroll
            for (int d = 0; d < D; ++d) {
                __builtin_nontemporal_store(lv, (v4f*)(leftout + o));
                o += HW;
            }
        }
    }

    // Wait for this wave's async copies, then sync the workgroup.
    asm volatile("s_wait_asynccnt 0" ::: "memory");
    __syncthreads();

    // ---- Disparity warp: gather from LDS, stream b128 NT stores ----
    const size_t dbase  = ((size_t)b * D * H + h) * W + wb;           // + d*HW
    const size_t obase0 = (size_t)b * C * D * HW
                        + (size_t)(cg * 8) * D * HW
                        + (size_t)h * W + wb;                          // + c*D*HW + d*HW

#pragma unroll 2
    for (int d = 0; d < D; ++d) {
        const v4f dv = __builtin_nontemporal_load(
            (const v4f*)(disp + dbase + (size_t)d * HW));
        int   ix[4];
        float vm[4];
#pragma unroll
        for (int j = 0; j < 4; ++j) {
            const float r  = (float)(wb + j) - dv[j];
            // clamp then truncate toward zero (jnp.clip(...).astype(int32))
            const float rc = fminf(fmaxf(r, 0.0f), (float)(W - 1));
            ix[j] = (int)rc;
            // mask uses the unclamped coordinate (reference: warped * valid)
            vm[j] = (r >= 0.0f && r <= (float)(W - 1)) ? 1.0f : 0.0f;
        }

        size_t o = obase0 + (size_t)d * HW;
#pragma unroll
        for (int k = 0; k < 8; ++k) {
            const float* srow = &smem[(cg * 8 + k) * W];
            v4f v;
            v[0] = srow[ix[0]] * vm[0];
            v[1] = srow[ix[1]] * vm[1];
            v[2] = srow[ix[2]] * vm[2];
            v[3] = srow[ix[3]] * vm[3];
            __builtin_nontemporal_store(v, (v4f*)(warped + o));
            o += (size_t)D * HW;
        }
    }
}

extern "C" void kernel_launch(void* const* d_in, const int* in_sizes, int n_in,
                              void* d_out, int out_size, void* d_ws, size_t ws_size,
                              hipStream_t stream) {
    (void)in_sizes; (void)n_in; (void)out_size; (void)d_ws; (void)ws_size;

    // setup_inputs order: left_input, right_input, disparity_samples (all f32)
    const float* left  = (const float*)d_in[0];
    const float* right = (const float*)d_in[1];
    const float* disp  = (const float*)d_in[2];

    // Reference returns (warped, left_feature_map): concatenated flat in d_out.
    float* warped  = (float*)d_out;
    float* leftout = (float*)d_out + (size_t)B * C * D * H * W;

    dim3 grid(B * H);   // 384 blocks
    dim3 block(384);    // 12 wave32s

    disparity_warp_kernel<<<grid, block, 0, stream>>>(left, right, disp,
                                                      warped, leftout);
}